// BiDecoder_80315888435313
// MI455X (gfx1250) — compile-verified
//
#include <hip/hip_runtime.h>

typedef __attribute__((ext_vector_type(2))) float v2f;
typedef __attribute__((ext_vector_type(4))) float v4f;
typedef __attribute__((ext_vector_type(8))) float v8f;

#define DIM        64
#define NREL       5
#define WAVES      4          // waves per block
#define EPT        32         // edges per wave-tile (two 16-row sub-tiles sharing B frags)
#define MAX_BLOCKS 2048       // persistent blocks (grid-stride over tiles)

// out[e,r] = sum_n ( (U W[r]^T)[e,n] + b[r][n] ) * I[e,n]
// Stage 1: Y = U @ W[r]^T + ones*b[r]  via v_wmma_f32_16x16x4_f32, M=32 as two 16-row
//          sub-tiles that SHARE each B fragment (1 ds_load_b64 -> 2 WMMAs).
// Stage 2: per sub-tile D2 = Y @ I^T, take diagonal.
//
// f32 WMMA fragment layouts (ISA 7.12.2):
//   A 16x4 : lane l, vgpr v -> (M=l%16, K=4s + 2*(l/16) + v)   => contiguous float2
//   B 4x16 : lane l, vgpr v -> (K=4s + 2*(l/16) + v, N=l%16)   => contiguous float2 of row-major W[r][n][*]
//   C/D    : lane l, vgpr v -> (M=v + 8*(l/16),   N=l%16)

__global__ __launch_bounds__(WAVES * 32)
void bidecoder_kernel(const float* __restrict__ u_feat,
                      const float* __restrict__ i_feat,
                      const int*   __restrict__ edge_user,
                      const int*   __restrict__ edge_item,
                      const float* __restrict__ W,     // [5][64][64]
                      const float* __restrict__ bias,  // [5][64]
                      float* __restrict__ out,         // [E][5]
                      int num_edges)
{
    __shared__ __align__(16) float lds_W [NREL * DIM * DIM];   // 80 KB, all relations, loaded once
    __shared__ __align__(16) float lds_b [NREL * DIM];         // 1.25 KB
    __shared__ __align__(16) float lds_U [WAVES][EPT * DIM];   // 32 KB (reused as Y in stage 2)
    __shared__ __align__(16) float lds_I [WAVES][EPT * DIM];   // 32 KB
    __shared__ __align__(16) float lds_O [WAVES][EPT * NREL];  // 2.5 KB

    const int tid  = threadIdx.x;
    const int wave = tid >> 5;
    const int lane = tid & 31;
    const int h    = lane >> 4;    // half-wave
    const int lm   = lane & 15;    // lane % 16

    // ---- stage ALL of W and b into LDS once per (persistent) block ----
    {
        const v4f* src = (const v4f*)W;
        for (int idx = tid; idx < (NREL * DIM * DIM) / 4; idx += WAVES * 32)
            ((v4f*)lds_W)[idx] = src[idx];
        for (int idx = tid; idx < NREL * DIM; idx += WAVES * 32)
            lds_b[idx] = bias[idx];
    }
    __syncthreads();               // the only barrier; waves independent afterwards

    float* Ubuf = lds_U[wave];     // doubles as Y buffer in stage 2
    float* Ibuf = lds_I[wave];
    float* Obuf = lds_O[wave];

    const int  tiles   = (num_edges + EPT - 1) / EPT;
    const long wstride = (long)gridDim.x * WAVES;

    for (long tile = (long)blockIdx.x * WAVES + wave; tile < tiles; tile += wstride) {
        const long e0 = tile * EPT;

        // ---- prefetch next tile's 64 feature rows (global_prefetch_b8) ----
        {
            long nt = tile + wstride;
            if (nt < tiles) {
                long epf = nt * EPT + lane;              // one U row + one I row per lane
                if (epf >= num_edges) epf = num_edges - 1;
                const float* pu = u_feat + (long)edge_user[epf] * DIM;   // 256 B rows
                const float* pi = i_feat + (long)edge_item[epf] * DIM;
                __builtin_prefetch(pu,      0, 3);
                __builtin_prefetch(pu + 32, 0, 3);
                __builtin_prefetch(pi,      0, 3);
                __builtin_prefetch(pi + 32, 0, 3);
            }
        }

        // ---- gather 32 U rows + 32 I rows into LDS (b128, coalesced) ----
        #pragma unroll
        for (int j = 0; j < (EPT * DIM) / (32 * 4); ++j) {   // 16 iterations
            int flat = j * 32 + lane;                        // 0..511
            int row  = flat >> 4;
            int c4   = (flat & 15) << 2;
            long e   = e0 + row;
            if (e >= num_edges) e = num_edges - 1;
            long ue = edge_user[e];
            long ie = edge_item[e];
            *(v4f*)(Ubuf + row * DIM + c4) = *(const v4f*)(u_feat + ue * DIM + c4);
            *(v4f*)(Ibuf + row * DIM + c4) = *(const v4f*)(i_feat + ie * DIM + c4);
        }
        asm volatile("" ::: "memory");   // same-wave LDS ops are in-order (HW); stop compiler reordering

        // ---- A fragments of both 16-row U sub-tiles held in registers ----
        v2f a0[16], a1[16];
        #pragma unroll
        for (int s = 0; s < 16; ++s) {
            a0[s] = *(const v2f*)(Ubuf + lm * DIM        + 4 * s + 2 * h);
            a1[s] = *(const v2f*)(Ubuf + (16 + lm) * DIM + 4 * s + 2 * h);
        }

        v2f a_one;                       // ones-column A fragment for bias k-step
        a_one.x = (h == 0) ? 1.0f : 0.0f;
        a_one.y = 0.0f;

        for (int r = 0; r < NREL; ++r) {
            const float* Wr = lds_W + r * DIM * DIM;

            // ---- stage 1: Y = U @ W[r]^T + b[r]  (32x64, four 16-wide N tiles) ----
            for (int t = 0; t < 4; ++t) {
                const int n = t * 16 + lm;
                v8f acc0 = {}, acc1 = {};
                #pragma unroll
                for (int s = 0; s < 16; ++s) {
                    v2f bf = *(const v2f*)(Wr + n * DIM + 4 * s + 2 * h);  // shared B frag
                    acc0 = __builtin_amdgcn_wmma_f32_16x16x4_f32(
                               false, a0[s], false, bf, (short)0, acc0, false, false);
                    acc1 = __builtin_amdgcn_wmma_f32_16x16x4_f32(
                               false, a1[s], false, bf, (short)0, acc1, false, false);
                }
                v2f bb;                  // bias k-step: B[64,n]=b[r][n], rest 0
                bb.x = (h == 0) ? lds_b[r * DIM + n] : 0.0f;
                bb.y = 0.0f;
                acc0 = __builtin_amdgcn_wmma_f32_16x16x4_f32(
                           false, a_one, false, bb, (short)0, acc0, false, false);
                acc1 = __builtin_amdgcn_wmma_f32_16x16x4_f32(
                           false, a_one, false, bb, (short)0, acc1, false, false);
                // D layout -> Y buffer (reusing Ubuf; A frags already live in regs)
                #pragma unroll
                for (int v = 0; v < 8; ++v) {
                    Ubuf[(v + 8 * h) * DIM + n]        = acc0[v];
                    Ubuf[(16 + v + 8 * h) * DIM + n]   = acc1[v];
                }
            }
            asm volatile("" ::: "memory");

            // ---- stage 2: per sub-tile D2 = Y @ I^T ; diagonal = per-edge dot ----
            #pragma unroll
            for (int p = 0; p < 2; ++p) {
                const float* Yp = Ubuf + (p * 16) * DIM;
                const float* Ip = Ibuf + (p * 16) * DIM;
                v8f acc2 = {};
                #pragma unroll
                for (int s = 0; s < 16; ++s) {
                    v2f af = *(const v2f*)(Yp + lm * DIM + 4 * s + 2 * h);  // Y row lm
                    v2f bf = *(const v2f*)(Ip + lm * DIM + 4 * s + 2 * h);  // B2[k,n]=I[n,k]
                    acc2 = __builtin_amdgcn_wmma_f32_16x16x4_f32(
                               false, af, false, bf, (short)0, acc2, false, false);
                }
                // diag(m==n): lanes 0-7 at vgpr=lane ; lanes 24-31 at vgpr=lane-24
                int vsel = (lane < 16) ? lane : (lane - 24);
                float dval = acc2[0];
                #pragma unroll
                for (int v = 1; v < 8; ++v) dval = (vsel == v) ? acc2[v] : dval;
                bool valid = (lane < 8) || (lane >= 24);
                int  m     = p * 16 + ((lane < 8) ? lane : (lane - 16));
                if (valid) Obuf[m * NREL + r] = dval;   // divergence ends before next WMMA
            }
            asm volatile("" ::: "memory");
        }

        // ---- store 32 edges x 5 relations, contiguous ----
        for (int j = lane; j < EPT * NREL; j += 32) {
            long o = e0 * NREL + j;
            if (o < (long)num_edges * NREL) out[o] = Obuf[j];
        }
    }
}

extern "C" void kernel_launch(void* const* d_in, const int* in_sizes, int n_in,
                              void* d_out, int out_size, void* d_ws, size_t ws_size,
                              hipStream_t stream) {
    const float* u_feat = (const float*)d_in[0];
    const float* i_feat = (const float*)d_in[1];
    const int*   eu     = (const int*)d_in[2];
    const int*   ei     = (const int*)d_in[3];
    const float* W      = (const float*)d_in[4];
    const float* b      = (const float*)d_in[5];
    float*       out    = (float*)d_out;

    int E      = in_sizes[2];
    int tiles  = (E + EPT - 1) / EPT;
    int blocks = (tiles + WAVES - 1) / WAVES;
    if (blocks > MAX_BLOCKS) blocks = MAX_BLOCKS;

    bidecoder_kernel<<<blocks, WAVES * 32, 0, stream>>>(u_feat, i_feat, eu, ei, W, b, out, E);
}